// LightDeepFilteringModule_91293824843922
// MI455X (gfx1250) — compile-verified
//
#include <hip/hip_runtime.h>
#include <stdint.h>

#define NF     96      // NUM_FREQS
#define FS     5       // FRAME_SIZE (taps)
#define T_DIM  1000
#define F_DIM  481

typedef unsigned int v4u __attribute__((ext_vector_type(4)));
typedef int          v8i __attribute__((ext_vector_type(8)));
typedef int          v4i __attribute__((ext_vector_type(4)));

__global__ __launch_bounds__(256) void df_fir_kernel(const float* __restrict__ spec,
                                                     const float* __restrict__ coefs,
                                                     float* __restrict__ out)
{
    // Coefs tile for this (b,t): 5 taps x 96 complex = 960 floats = 3840 B
    __shared__ __align__(16) float lds_c[FS * NF * 2];

    const int bt  = blockIdx.x;
    const int b   = bt / T_DIM;
    const int t   = bt - b * T_DIM;
    const int tid = threadIdx.x;

    const size_t row_f = (size_t)bt * (F_DIM * 2);             // floats
    const float2* __restrict__ srow = (const float2*)(spec + row_f);
    float2* __restrict__       orow = (float2*)(out + row_f);

    // Byte address of coefs[b, 0, t, 0, 0]
    const uint64_t coef_ga =
        (uint64_t)(uintptr_t)(coefs + ((size_t)(b * FS) * T_DIM + t) * (NF * 2));

#if __has_builtin(__builtin_amdgcn_tensor_load_to_lds)
    if ((tid >> 5) == 0) {                 // wave 0 issues the TDM descriptor
        const uint32_t lds_addr = (uint32_t)(uintptr_t)(&lds_c[0]);
        // ---- D# group 0 (128b): count=1, lds_addr, global_addr[56:0], type=2
        v4u g0;
        g0.x = 0x1u;                                              // count=1, user mode
        g0.y = lds_addr;                                          // LDS byte address
        g0.z = (uint32_t)(coef_ga & 0xFFFFFFFFu);                 // global_addr[31:0]
        g0.w = (uint32_t)((coef_ga >> 32) & 0x01FFFFFFu)          // global_addr[56:32]
             | (0x2u << 30);                                      // type = 2 ("image")
        // ---- D# group 1 (256b): 2D tile, 4-byte elements
        v8i g1;
        g1[0] = (int)(2u << 16);                 // wg_mask=0 (not in cluster), data_size=4B
        g1[1] = (int)((uint32_t)(NF * 2) << 16); // atomic_barrier=0 | tensor_dim0.lo16 = 192
        g1[2] = (int)((uint32_t)FS << 16);       // tensor_dim0.hi16=0 | tensor_dim1.lo16 = 5
        g1[3] = (int)((uint32_t)(NF * 2) << 16); // tensor_dim1.hi16=0 | tile_dim0 = 192
        g1[4] = FS;                              // tile_dim1 = 5, tile_dim2 = 0 (unused)
        g1[5] = T_DIM * NF * 2;                  // tensor_dim0_stride.lo32 = 192000
        g1[6] = 0;                               // stride0.hi16 = 0, stride1.lo16 = 0
        g1[7] = 0;
        v4i g2 = {0, 0, 0, 0};                   // dims 2/3 unused (2D tensor)
        v4i g3 = {0, 0, 0, 0};
        v8i g4 = {0, 0, 0, 0, 0, 0, 0, 0};       // unused trailing group (6-arg form)
        __builtin_amdgcn_tensor_load_to_lds(g0, g1, g2, g3, g4, 0);
    }
#else
    // Fallback: coalesced staging of the coef tile
    for (int i = tid; i < FS * NF * 2; i += 256) {
        const int n = i / (NF * 2);
        const int j = i - n * (NF * 2);
        lds_c[i] = coefs[(((size_t)(b * FS + n)) * T_DIM + t) * (NF * 2) + j];
    }
#endif

    // Pass-through for f in [96, 481) while the coef tile is in flight.
    // Rows are 8B-aligned (962 floats/row), so float2 is the widest safe vector.
    for (int j = tid; j < (F_DIM - NF); j += 256) {
        orow[NF + j] = srow[NF + j];
    }

#if __has_builtin(__builtin_amdgcn_tensor_load_to_lds)
    if ((tid >> 5) == 0) {
  #if __has_builtin(__builtin_amdgcn_s_wait_tensorcnt)
        __builtin_amdgcn_s_wait_tensorcnt(0);
  #else
        asm volatile("s_wait_tensorcnt 0x0" ::: "memory");
  #endif
    }
#endif
    __syncthreads();

    // 5-tap complex FIR for f in [0, 96): lanes 0..95 (waves 0-2)
    if (tid < NF) {
        const int f = tid;
        float re = 0.f, im = 0.f;
#pragma unroll
        for (int n = 0; n < FS; ++n) {
            const int ts = t + n - (FS - 1);     // zero-pad at the front (LOOKAHEAD=0)
            if (ts >= 0) {
                const float2 s =
                    ((const float2*)(spec + ((size_t)(b * T_DIM + ts)) * (F_DIM * 2)))[f];
                const float cr = lds_c[n * (NF * 2) + 2 * f];
                const float ci = lds_c[n * (NF * 2) + 2 * f + 1];
                re = fmaf(s.x, cr, fmaf(-s.y, ci, re));   // Re(s*c)
                im = fmaf(s.x, ci, fmaf(s.y, cr, im));    // Im(s*c)
            }
        }
        orow[f] = make_float2(re, im);
    }
}

extern "C" void kernel_launch(void* const* d_in, const int* in_sizes, int n_in,
                              void* d_out, int out_size, void* d_ws, size_t ws_size,
                              hipStream_t stream)
{
    const float* spec  = (const float*)d_in[0];
    const float* coefs = (const float*)d_in[1];
    float* out = (float*)d_out;

    const int B = in_sizes[0] / (T_DIM * F_DIM * 2);   // 32
    dim3 grid((unsigned)(B * T_DIM));                  // one block per (b, t) row
    df_fir_kernel<<<grid, 256, 0, stream>>>(spec, coefs, out);
}